// Haar_87402584473886
// MI455X (gfx1250) — compile-verified
//
#include <hip/hip_runtime.h>
#include <hip/hip_bf16.h>

// Haar wavelet step, MI455X (gfx1250).
// x:(64,4096,128) f32 -> approx:(64,2048,128), r scalar, details:(64,128,2048)
// Memory-bound: ~268 MB HBM traffic -> ~11.5us roofline at 23.3 TB/s.
// Main kernel: b128 streaming loads/stores + LDS-staged transpose (stride 129,
// conflict-free transposed reads). Scalar reduction finished with
// V_WMMA_F32_16X16X4_F32 (exact f32 row-sum via ones matrix).

#define INV_SQRT2F 0.70710678118654752440f
#define B_DIM   64
#define T_DIM   4096
#define T2_DIM  2048
#define C_DIM   128
#define APPROX_N 16777216   // 64*2048*128
#define TILE_T  64          // t-values per block
#define LDS_STRIDE 129      // 128 + 1 pad: conflict-free transposed reads

typedef float v2f __attribute__((ext_vector_type(2)));
typedef float v4f __attribute__((ext_vector_type(4)));   // clang vector: OK for nontemporal builtins
typedef float v8f __attribute__((ext_vector_type(8)));

__global__ __launch_bounds__(256) void haar_main_kernel(
    const float* __restrict__ x,
    float* __restrict__ approx,
    float* __restrict__ details,     // 4B-aligned only (odd flat offset)
    float2* __restrict__ partials)
{
    __shared__ float det[TILE_T * LDS_STRIDE];   // 33,024 B
    __shared__ float2 wred[8];

    const int tid = threadIdx.x;
    const int t0  = blockIdx.x * TILE_T;         // blockIdx.x in [0,32)
    const int b   = blockIdx.y;                  // [0,64)

    const float* xb = x      + (size_t)b * T_DIM  * C_DIM;
    float*       ab = approx + (size_t)b * T2_DIM * C_DIM;

    float sx  = 0.0f;   // sum of all x elements seen by this thread
    float sad = 0.0f;   // sum of |details| elements

    // Phase 1: load pairs, compute approx (direct store) + details (to LDS).
    // li in [0,2048): c4 = float4 column (32 per row), t = tile row.
    // A wave covers one full 512B row of C -> perfectly coalesced b128.
#pragma unroll
    for (int i = 0; i < 8; ++i) {
        const int li = i * 256 + tid;
        const int c4 = li & 31;
        const int t  = li >> 5;
        const int tg = t0 + t;

        const v4f* pe = (const v4f*)(xb + (size_t)(2 * tg)     * C_DIM) + c4;
        const v4f* po = (const v4f*)(xb + (size_t)(2 * tg + 1) * C_DIM) + c4;
        const v4f e = __builtin_nontemporal_load(pe);
        const v4f o = __builtin_nontemporal_load(po);

        const v4f a = (e + o) * INV_SQRT2F;
        const v4f d = (e - o) * INV_SQRT2F;

        sx  += (e[0] + e[1] + e[2] + e[3]) + (o[0] + o[1] + o[2] + o[3]);
        sad += fabsf(d[0]) + fabsf(d[1]) + fabsf(d[2]) + fabsf(d[3]);

        __builtin_nontemporal_store(a, (v4f*)(ab + (size_t)tg * C_DIM) + c4);

        const int base = t * LDS_STRIDE + c4 * 4;
        det[base + 0] = d[0];
        det[base + 1] = d[1];
        det[base + 2] = d[2];
        det[base + 3] = d[3];
    }

    // Per-wave reduction of the two partial sums (wave32).
#pragma unroll
    for (int off = 16; off > 0; off >>= 1) {
        sx  += __shfl_xor(sx,  off, 32);
        sad += __shfl_xor(sad, off, 32);
    }
    if ((tid & 31) == 0) wred[tid >> 5] = make_float2(sx, sad);

    __syncthreads();

    // Phase 2: transposed write-out of details from LDS.
    // lo in [0,8192): t fast (coalesced 128B scalar stores), c slow.
    // LDS read bank = (t + c) mod 64 across a wave -> conflict-free.
    float* db = details + (size_t)b * C_DIM * T2_DIM + t0;
#pragma unroll
    for (int i = 0; i < 32; ++i) {
        const int lo = i * 256 + tid;
        const int t  = lo & 63;
        const int c  = lo >> 6;
        __builtin_nontemporal_store(det[t * LDS_STRIDE + c],
                                    db + (size_t)c * T2_DIM + t);
    }

    if (tid == 0) {
        float2 acc = wred[0];
#pragma unroll
        for (int w = 1; w < 8; ++w) { acc.x += wred[w].x; acc.y += wred[w].y; }
        partials[blockIdx.y * gridDim.x + blockIdx.x] = acc;
    }
}

// Reduce 2048 block partials; final 32-lane fold done with
// V_WMMA_F32_16X16X4_F32: D = A(16x4) x ones(4x16). A-layout puts lane m's
// v2f into row m (K=0,1) and lane m+16's into row m (K=2,3), so row-sum m =
// val_m + val_{m+16}; D rows 0-7 land in lanes 0-15, rows 8-15 in lanes 16-31.
__global__ __launch_bounds__(32) void haar_finish_kernel(
    const float2* __restrict__ partials,
    float* __restrict__ r_out)
{
    const int lane = threadIdx.x;
    float sx = 0.0f, sad = 0.0f;
    for (int i = lane; i < 2048; i += 32) {
        const float2 p = partials[i];
        sx += p.x; sad += p.y;
    }

    v2f ones; ones[0] = 1.0f; ones[1] = 1.0f;
    v8f zero = {};
    v2f ax;  ax[0] = sx;  ax[1] = 0.0f;
    v2f ad;  ad[0] = sad; ad[1] = 0.0f;

    // 8 args: (neg_a, A, neg_b, B, c_mod, C, reuse_a, reuse_b)
    v8f dx = __builtin_amdgcn_wmma_f32_16x16x4_f32(
        false, ax, false, ones, (short)0, zero, false, false);
    v8f dd = __builtin_amdgcn_wmma_f32_16x16x4_f32(
        false, ad, false, ones, (short)0, zero, false, false);

    float tx = 0.0f, td = 0.0f;
#pragma unroll
    for (int j = 0; j < 8; ++j) { tx += dx[j]; td += dd[j]; }
    // lanes 0-15 hold sum of rows 0-7; lanes 16-31 hold sum of rows 8-15.
    const float SX  = __shfl(tx, 0, 32) + __shfl(tx, 16, 32);
    const float SAD = __shfl(td, 0, 32) + __shfl(td, 16, 32);

    if (lane == 0) {
        const float n_half = (float)APPROX_N;            // elements in approx/details
        const float rd     = 0.01f * (SAD / n_half);
        const float mean_x = SX / (2.0f * n_half);
        const float mean_a = (SX * INV_SQRT2F) / n_half; // sum(approx)=sum(x)/sqrt2
        const float rc     = 0.01f * fabsf(mean_a - mean_x);
        r_out[0] = rd + rc;
    }
}

extern "C" void kernel_launch(void* const* d_in, const int* in_sizes, int n_in,
                              void* d_out, int out_size, void* d_ws, size_t ws_size,
                              hipStream_t stream) {
    const float* x = (const float*)d_in[0];
    float* out = (float*)d_out;

    float*  approx  = out;                       // [0, 16777216)
    float*  r       = out + APPROX_N;            // [16777216]
    float*  details = out + APPROX_N + 1;        // [16777217, 33554433)
    float2* partials = (float2*)d_ws;            // 2048 * 8 B = 16 KB

    dim3 grid(T2_DIM / TILE_T, B_DIM);           // (32, 64) = 2048 blocks
    haar_main_kernel<<<grid, 256, 0, stream>>>(x, approx, details, partials);
    haar_finish_kernel<<<1, 32, 0, stream>>>(partials, r);
}